// BertStackSegmentor_44186623542046
// MI455X (gfx1250) — compile-verified
//
#include <hip/hip_runtime.h>
#include <hip/hip_bf16.h>

// ---------------- model constants ----------------
#define BB 64
#define TT 256
#define HH 768
#define H2 (2 * HH)      // 1536
#define GH (4 * HH)      // 3072 gates width
#define STK 257          // stack slots (max pointer 255, push at 256)

typedef __attribute__((ext_vector_type(2))) float v2f;
typedef __attribute__((ext_vector_type(8))) float v8f;

__device__ __forceinline__ float sigf(float x) { return 1.0f / (1.0f + expf(-x)); }

__device__ __forceinline__ float waveReduce(float s) {
#pragma unroll
  for (int o = 16; o > 0; o >>= 1) s += __shfl_down(s, o, 32);
  return s;
}

// ---------------------------------------------------------------------------
// Two-segment WMMA f32 GEMM with fixed M=64:
//   C[64,N] = A0[64,K0] @ W0[N,K0]^T + A1[64,K1] @ W1[N,K1]^T + bias[N]
// One wave owns a 64x16 output strip (4 accumulator tiles) so each B
// fragment is loaded once and fed to 4 v_wmma_f32_16x16x4_f32 ops: the
// weight stream (dominant L2 traffic) is read exactly once per GEMM.
// A-frag: lane(0-15): M=m0+lane, K={kb,kb+1}; lane(16-31): K={kb+2,kb+3}
// B-frag: lane(0-15): N=n0+lane, K={kb,kb+1}; lane(16-31): K={kb+2,kb+3}
// C/D:    vgpr r -> (M = m0 + r + 8*(lane>>4), N = n0 + (lane&15))
// ---------------------------------------------------------------------------
__global__ void gemm2seg_wmma_f32(const float* __restrict__ A0, long lda0,
                                  const float* __restrict__ W0, int K0,
                                  const float* __restrict__ A1, long lda1,
                                  const float* __restrict__ W1, int K1,
                                  const float* __restrict__ bias,
                                  float* __restrict__ C, int ldc, int N) {
  const int lane = threadIdx.x & 31;
  const int wave = threadIdx.x >> 5;
  const int nt = blockIdx.x * (blockDim.x >> 5) + wave;  // N-tile index
  const int n0 = nt * 16;
  if (n0 >= N) return;  // whole-wave uniform exit
  const int lo = lane & 15;
  const int hi = lane >> 4;
  const int ncol = n0 + lo;

  v8f acc0 = {}, acc1 = {}, acc2 = {}, acc3 = {};

  {
    const float* ap0 = A0 + (long)(lo +  0) * lda0 + 2 * hi;
    const float* ap1 = A0 + (long)(lo + 16) * lda0 + 2 * hi;
    const float* ap2 = A0 + (long)(lo + 32) * lda0 + 2 * hi;
    const float* ap3 = A0 + (long)(lo + 48) * lda0 + 2 * hi;
    const float* wp  = W0 + (long)ncol * (long)K0 + 2 * hi;
#pragma unroll 4
    for (int k = 0; k < K0; k += 4) {
      v2f b  = *(const v2f*)(wp + k);
      v2f a0 = *(const v2f*)(ap0 + k);
      v2f a1 = *(const v2f*)(ap1 + k);
      v2f a2 = *(const v2f*)(ap2 + k);
      v2f a3 = *(const v2f*)(ap3 + k);
      acc0 = __builtin_amdgcn_wmma_f32_16x16x4_f32(false, a0, false, b, (short)0, acc0, false, false);
      acc1 = __builtin_amdgcn_wmma_f32_16x16x4_f32(false, a1, false, b, (short)0, acc1, false, false);
      acc2 = __builtin_amdgcn_wmma_f32_16x16x4_f32(false, a2, false, b, (short)0, acc2, false, false);
      acc3 = __builtin_amdgcn_wmma_f32_16x16x4_f32(false, a3, false, b, (short)0, acc3, false, false);
    }
  }
  if (K1 > 0) {
    const float* ap0 = A1 + (long)(lo +  0) * lda1 + 2 * hi;
    const float* ap1 = A1 + (long)(lo + 16) * lda1 + 2 * hi;
    const float* ap2 = A1 + (long)(lo + 32) * lda1 + 2 * hi;
    const float* ap3 = A1 + (long)(lo + 48) * lda1 + 2 * hi;
    const float* wp  = W1 + (long)ncol * (long)K1 + 2 * hi;
#pragma unroll 4
    for (int k = 0; k < K1; k += 4) {
      v2f b  = *(const v2f*)(wp + k);
      v2f a0 = *(const v2f*)(ap0 + k);
      v2f a1 = *(const v2f*)(ap1 + k);
      v2f a2 = *(const v2f*)(ap2 + k);
      v2f a3 = *(const v2f*)(ap3 + k);
      acc0 = __builtin_amdgcn_wmma_f32_16x16x4_f32(false, a0, false, b, (short)0, acc0, false, false);
      acc1 = __builtin_amdgcn_wmma_f32_16x16x4_f32(false, a1, false, b, (short)0, acc1, false, false);
      acc2 = __builtin_amdgcn_wmma_f32_16x16x4_f32(false, a2, false, b, (short)0, acc2, false, false);
      acc3 = __builtin_amdgcn_wmma_f32_16x16x4_f32(false, a3, false, b, (short)0, acc3, false, false);
    }
  }

  const float bv = bias ? bias[ncol] : 0.0f;
#pragma unroll
  for (int r = 0; r < 8; ++r) {
    const int row = r + 8 * hi;
    C[(long)(row +  0) * ldc + ncol] = acc0[r] + bv;
    C[(long)(row + 16) * ldc + ncol] = acc1[r] + bv;
    C[(long)(row + 32) * ldc + ncol] = acc2[r] + bv;
    C[(long)(row + 48) * ldc + ncol] = acc3[r] + bv;
  }
}

// ---------------------------------------------------------------------------
// Bi-LSTM cell (one direction, one time step).  gates layout [B, 4H] = i,f,g,o
// ---------------------------------------------------------------------------
__global__ void lstm_cell_seq(const float* __restrict__ gates,
                              float* __restrict__ h, float* __restrict__ c,
                              float* __restrict__ lstm_out, int t, int dirOff) {
  int i = blockIdx.x * blockDim.x + threadIdx.x;
  if (i >= BB * HH) return;
  int b = i / HH, j = i - b * HH;
  const float* g = gates + (long)b * GH;
  float gi = g[j], gf = g[HH + j], gg = g[2 * HH + j], go = g[3 * HH + j];
  float cn = sigf(gf) * c[i] + sigf(gi) * tanhf(gg);
  float hn = sigf(go) * tanhf(cn);
  c[i] = cn;
  h[i] = hn;
  lstm_out[((long)b * TT + t) * H2 + dirOff + j] = hn;
}

// gather top-of-stack h into contiguous [B,H] buffer for the recurrent GEMM
__global__ void gather_top(const float* __restrict__ stack,
                           const int* __restrict__ pos,
                           float* __restrict__ dst) {
  int i = blockIdx.x * blockDim.x + threadIdx.x;
  if (i >= BB * HH) return;
  int b = i / HH, j = i - b * HH;
  dst[i] = stack[((long)pos[b] * BB + b) * HH + j];
}

// subword stack-LSTM cell: push (h1,c1) at spos+1, emit concat(h1,c1)
__global__ void stack_cell_subw(const float* __restrict__ gates,
                                float* __restrict__ sh, float* __restrict__ sc,
                                const int* __restrict__ spos,
                                float* __restrict__ subword) {
  int i = blockIdx.x * blockDim.x + threadIdx.x;
  if (i >= BB * HH) return;
  int b = i / HH, j = i - b * HH;
  int p = spos[b];
  const float* g = gates + (long)b * GH;
  float gi = g[j], gf = g[HH + j], gg = g[2 * HH + j], go = g[3 * HH + j];
  float ct = sc[((long)p * BB + b) * HH + j];
  float cn = sigf(gf) * ct + sigf(gi) * tanhf(gg);
  float hn = sigf(go) * tanhf(cn);
  long dst = ((long)(p + 1) * BB + b) * HH + j;
  sh[dst] = hn;
  sc[dst] = cn;
  subword[(long)b * H2 + j] = hn;
  subword[(long)b * H2 + HH + j] = cn;
}

// word stack-LSTM cell: push (h2,c2) at wpos+1, emit h2, advance stack pointers
__global__ void stack_cell_word(const float* __restrict__ gates,
                                float* __restrict__ wh, float* __restrict__ wc,
                                const int* __restrict__ wposCur,
                                float* __restrict__ h2buf,
                                const int* __restrict__ sposCur,
                                int* __restrict__ sposNxt,
                                int* __restrict__ wposNxt,
                                const int* __restrict__ golds, int t) {
  int i = blockIdx.x * blockDim.x + threadIdx.x;
  if (i >= BB * HH) return;
  int b = i / HH, j = i - b * HH;
  int p = wposCur[b];
  const float* g = gates + (long)b * GH;
  float gi = g[j], gf = g[HH + j], gg = g[2 * HH + j], go = g[3 * HH + j];
  float ct = wc[((long)p * BB + b) * HH + j];
  float cn = sigf(gf) * ct + sigf(gi) * tanhf(gg);
  float hn = sigf(go) * tanhf(cn);
  long dst = ((long)(p + 1) * BB + b) * HH + j;
  wh[dst] = hn;
  wc[dst] = cn;
  h2buf[i] = hn;
  if (j == 0) {
    int gv = golds[(long)b * TT + (t + 1)];
    // g==0 -> APP: push subword (spos+1), hold word; g==1 -> SEP: clear subword, push word
    sposNxt[b] = (gv == 0) ? (sposCur[b] + 1) : 0;
    wposNxt[b] = (gv == 1) ? (p + 1) : p;
  }
}

// precompute the x_cur part of the classifier: clsx[(b*(T-1)+t)*2+cls]
//   = lstm_out[b, t+1, :] . cls_W[cls, H:3H] + cls_b[cls]
__global__ void cls_x_pre(const float* __restrict__ lstm_out,
                          const float* __restrict__ clsW,
                          const float* __restrict__ clsb,
                          float* __restrict__ clsx) {
  int wid = (blockIdx.x * blockDim.x + threadIdx.x) >> 5;
  int lane = threadIdx.x & 31;
  if (wid >= BB * (TT - 1) * 2) return;
  int cls = wid & 1;
  int bt = wid >> 1;
  int b = bt / (TT - 1);
  int t = bt - b * (TT - 1);
  const float* x = lstm_out + ((long)b * TT + t + 1) * H2;
  const float* w = clsW + (long)cls * (3 * HH) + HH;
  float s = 0.0f;
  for (int k = lane; k < H2; k += 32) s += x[k] * w[k];
  s = waveReduce(s);
  if (lane == 0) clsx[wid] = s + clsb[cls];
}

// per-step classifier: out[b, t+1, cls] = h2 . cls_W[cls, 0:H] + clsx
__global__ void cls_step(const float* __restrict__ h2,
                         const float* __restrict__ clsW,
                         const float* __restrict__ clsx,
                         float* __restrict__ out, int t) {
  int wid = (blockIdx.x * blockDim.x + threadIdx.x) >> 5;
  int lane = threadIdx.x & 31;
  if (wid >= BB * 2) return;
  int cls = wid & 1;
  int b = wid >> 1;
  const float* w = clsW + (long)cls * (3 * HH);
  const float* hv = h2 + (long)b * HH;
  float s = 0.0f;
  for (int k = lane; k < HH; k += 32) s += hv[k] * w[k];
  s = waveReduce(s);
  if (lane == 0)
    out[((long)b * TT + t + 1) * 2 + cls] =
        s + clsx[((long)(b * (TT - 1) + t) << 1) | cls];
}

// zero the recurrent state, stack slot 0, stack pointers; write out[:,0,:]
__global__ void init_state(float* hf, float* cf, float* hb, float* cb,
                           float* sh, float* sc, float* wh, float* wc,
                           int* sposCur, int* wposCur, float* out) {
  int i = blockIdx.x * blockDim.x + threadIdx.x;
  if (i < BB * HH) {
    hf[i] = 0.f; cf[i] = 0.f; hb[i] = 0.f; cb[i] = 0.f;
    sh[i] = 0.f; sc[i] = 0.f; wh[i] = 0.f; wc[i] = 0.f;  // stack slot 0
  }
  if (i < BB) {
    sposCur[i] = 0;
    wposCur[i] = 0;
    out[(long)i * TT * 2 + 0] = -1.0f;
    out[(long)i * TT * 2 + 1] = 1.0f;
  }
}

// ---------------------------------------------------------------------------
extern "C" void kernel_launch(void* const* d_in, const int* in_sizes, int n_in,
                              void* d_out, int out_size, void* d_ws,
                              size_t ws_size, hipStream_t stream) {
  const float* hidden   = (const float*)d_in[0];   // [B,T,H]
  const int*   golds    = (const int*)d_in[1];     // [B,T]
  const float* Wih_f    = (const float*)d_in[2];   // [4H,H]
  const float* Whh_f    = (const float*)d_in[3];   // [4H,H]
  const float* b_f      = (const float*)d_in[4];   // [4H]
  const float* Wih_b    = (const float*)d_in[5];
  const float* Whh_b    = (const float*)d_in[6];
  const float* b_b      = (const float*)d_in[7];
  const float* subw_Wih = (const float*)d_in[8];   // [4H,2H]
  const float* subw_Whh = (const float*)d_in[9];   // [4H,H]
  const float* subw_b   = (const float*)d_in[10];
  const float* word_Wih = (const float*)d_in[11];  // [4H,2H]
  const float* word_Whh = (const float*)d_in[12];  // [4H,H]
  const float* word_b   = (const float*)d_in[13];
  const float* clsW     = (const float*)d_in[14];  // [2,3H]
  const float* clsb     = (const float*)d_in[15];  // [2]
  float* out = (float*)d_out;                      // [B,T,2]

  // ---- carve workspace (float units; all chunk sizes are even -> 8B aligned)
  float* ws = (float*)d_ws;
  size_t o = 0;
  float* lstm_out = ws + o; o += (size_t)BB * TT * H2;   // 25.2M
  float* sh = ws + o; o += (size_t)STK * BB * HH;        // 12.6M
  float* sc = ws + o; o += (size_t)STK * BB * HH;
  float* wh = ws + o; o += (size_t)STK * BB * HH;
  float* wc = ws + o; o += (size_t)STK * BB * HH;
  float* gates_f = ws + o; o += (size_t)BB * GH;
  float* gates_b = ws + o; o += (size_t)BB * GH;
  float* gates_s = ws + o; o += (size_t)BB * GH;
  float* h_f = ws + o; o += (size_t)BB * HH;
  float* c_f = ws + o; o += (size_t)BB * HH;
  float* h_b = ws + o; o += (size_t)BB * HH;
  float* c_b = ws + o; o += (size_t)BB * HH;
  float* htop = ws + o; o += (size_t)BB * HH;
  float* wtop = ws + o; o += (size_t)BB * HH;
  float* subword = ws + o; o += (size_t)BB * H2;
  float* h2buf = ws + o; o += (size_t)BB * HH;
  float* clsx = ws + o; o += (size_t)BB * (TT - 1) * 2;
  int* ip = (int*)(ws + o);
  int* spos = ip;            // [2][B]
  int* wpos = ip + 2 * BB;   // [2][B]

  const dim3 gblk(128);                 // 4 waves / block, each wave a 64x16 strip
  const dim3 ggrid(GH / 16 / 4, 1);     // 192 N-tiles / 4 waves = 48 blocks
  const int eblk = 256;
  const int egrd = (BB * HH + eblk - 1) / eblk;  // 192

  // ---- init
  init_state<<<egrd, eblk, 0, stream>>>(h_f, c_f, h_b, c_b, sh, sc, wh, wc,
                                        spos, wpos, out);

  // ---- phase 1: bidirectional LSTM (256 sequential steps, 2 directions)
  for (int t = 0; t < TT; ++t) {
    // forward dir, time t:  gates = x_t@Wih_f^T + h_f@Whh_f^T + b_f
    gemm2seg_wmma_f32<<<ggrid, gblk, 0, stream>>>(
        hidden + (long)t * HH, (long)TT * HH, Wih_f, HH,
        h_f, (long)HH, Whh_f, HH, b_f, gates_f, GH, GH);
    lstm_cell_seq<<<egrd, eblk, 0, stream>>>(gates_f, h_f, c_f, lstm_out, t, 0);

    // backward dir, time T-1-t
    const int tb = TT - 1 - t;
    gemm2seg_wmma_f32<<<ggrid, gblk, 0, stream>>>(
        hidden + (long)tb * HH, (long)TT * HH, Wih_b, HH,
        h_b, (long)HH, Whh_b, HH, b_b, gates_b, GH, GH);
    lstm_cell_seq<<<egrd, eblk, 0, stream>>>(gates_b, h_b, c_b, lstm_out, tb, HH);
  }

  // ---- precompute classifier x_cur contribution (big parallel dot products)
  {
    const int nwaves = BB * (TT - 1) * 2;
    const int blocks = (nwaves * 32 + eblk - 1) / eblk;
    cls_x_pre<<<blocks, eblk, 0, stream>>>(lstm_out, clsW, clsb, clsx);
  }

  // ---- phase 2: stack-LSTM scan (255 sequential steps)
  for (int t = 0; t < TT - 1; ++t) {
    const int cur = t & 1, nxt = cur ^ 1;
    const int* sposCur = spos + cur * BB;
    int*       sposNxt = spos + nxt * BB;
    const int* wposCur = wpos + cur * BB;
    int*       wposNxt = wpos + nxt * BB;

    // subword cell
    gather_top<<<egrd, eblk, 0, stream>>>(sh, sposCur, htop);
    gemm2seg_wmma_f32<<<ggrid, gblk, 0, stream>>>(
        lstm_out + (long)t * H2, (long)TT * H2, subw_Wih, H2,
        htop, (long)HH, subw_Whh, HH, subw_b, gates_s, GH, GH);
    stack_cell_subw<<<egrd, eblk, 0, stream>>>(gates_s, sh, sc, sposCur, subword);

    // word cell
    gather_top<<<egrd, eblk, 0, stream>>>(wh, wposCur, wtop);
    gemm2seg_wmma_f32<<<ggrid, gblk, 0, stream>>>(
        subword, (long)H2, word_Wih, H2,
        wtop, (long)HH, word_Whh, HH, word_b, gates_s, GH, GH);
    stack_cell_word<<<egrd, eblk, 0, stream>>>(gates_s, wh, wc, wposCur, h2buf,
                                               sposCur, sposNxt, wposNxt,
                                               golds, t);

    // classifier row for this step: 128 waves
    cls_step<<<16, 256, 0, stream>>>(h2buf, clsW, clsx, out, t);
  }
}